// SoftAttention_16758962389785
// MI455X (gfx1250) — compile-verified
//
#include <hip/hip_runtime.h>

typedef __attribute__((ext_vector_type(2))) float v2f;
typedef __attribute__((ext_vector_type(8))) float v8f;

#define C_DIM 256
#define N_DIM 16
#define M_DIM 8192
#define TM    16     // m-rows per block
#define CT    128    // output columns (of S/alpha/z) per block

// LDS: sS [N][TM][CT] = 16*16*128*4 = 128 KB ; sY [TM][C] = 16 KB  (total 144 KB)
__global__ __launch_bounds__(256, 1)
void soft_attention_fused(const float* __restrict__ g,
                          const float* __restrict__ h_prev,
                          const float* __restrict__ Wg,
                          const float* __restrict__ Wh,
                          const float* __restrict__ Wy,
                          const float* __restrict__ by,
                          float* __restrict__ alpha_out,
                          float* __restrict__ z_out)
{
    __shared__ float sS[N_DIM * TM * CT];
    __shared__ float sY[TM * C_DIM];

    const int tid  = threadIdx.x;
    const int lane = tid & 31;
    const int wave = tid >> 5;      // 0..7
    const int half = lane >> 4;     // 0/1 : lane half selects K-subgroup
    const int lh   = lane & 15;     // row (A) / col (B,C,D) index

    const int m0   = blockIdx.x * TM;   // m-row block
    const int cblk = blockIdx.y * CT;   // output-column block

    // ---------------- Step 1: H = h_prev[m0:m0+16] @ Wh^T (all 256 cols) ---------
    // wave owns Y/H column tiles {wave, wave+8} (16 cols each)
    v8f Hacc0 = {0.f,0.f,0.f,0.f,0.f,0.f,0.f,0.f};
    v8f Hacc1 = Hacc0;
    {
        const float* Arow = h_prev + (size_t)(m0 + lh) * C_DIM + 2*half;
        const float* B0   = Wh + (size_t)(wave*16       + lh) * C_DIM + 2*half;
        const float* B1   = Wh + (size_t)((wave+8)*16   + lh) * C_DIM + 2*half;
        __builtin_prefetch(B0, 0, 1);
        __builtin_prefetch(B1, 0, 1);
        #pragma unroll 8
        for (int k0 = 0; k0 < C_DIM; k0 += 4) {
            v2f a  = *(const v2f*)(Arow + k0);
            v2f b0 = *(const v2f*)(B0 + k0);
            v2f b1 = *(const v2f*)(B1 + k0);
            Hacc0 = __builtin_amdgcn_wmma_f32_16x16x4_f32(false, a, false, b0, (short)0, Hacc0, false, false);
            Hacc1 = __builtin_amdgcn_wmma_f32_16x16x4_f32(false, a, false, b1, (short)0, Hacc1, false, false);
        }
    }

    // bias value for this lane's S column (wave owns S column tile == wave)
    const float bval = by[cblk + wave*16 + lh];

    // ---------------- Step 2: per glimpse i: Y -> tanh -> LDS -> S -> LDS --------
    for (int i = 0; i < N_DIM; ++i) {
        // Y tile = g[i, m-rows] @ Wg^T + H  (all 256 cols, 2 tiles per wave)
        v8f Y0 = Hacc0, Y1 = Hacc1;
        {
            const float* Arow = g  + ((size_t)i * M_DIM + m0 + lh) * C_DIM + 2*half;
            const float* B0   = Wg + (size_t)(wave*16     + lh) * C_DIM + 2*half;
            const float* B1   = Wg + (size_t)((wave+8)*16 + lh) * C_DIM + 2*half;
            #pragma unroll 8
            for (int k0 = 0; k0 < C_DIM; k0 += 4) {
                v2f a  = *(const v2f*)(Arow + k0);
                v2f b0 = *(const v2f*)(B0 + k0);
                v2f b1 = *(const v2f*)(B1 + k0);
                Y0 = __builtin_amdgcn_wmma_f32_16x16x4_f32(false, a, false, b0, (short)0, Y0, false, false);
                Y1 = __builtin_amdgcn_wmma_f32_16x16x4_f32(false, a, false, b1, (short)0, Y1, false, false);
            }
        }
        __syncthreads();   // previous iteration done reading sY
        // tanh + store Y tile: C/D element (row j+8*half, col tile*16+lh)
        #pragma unroll
        for (int j = 0; j < 8; ++j) {
            sY[(j + 8*half) * C_DIM + wave*16     + lh] = tanhf(Y0[j]);
            sY[(j + 8*half) * C_DIM + (wave+8)*16 + lh] = tanhf(Y1[j]);
        }
        __syncthreads();   // sY complete for all 256 cols

        // S tile = Y @ Wy^T + by  (wave owns one 16-col tile of the CT cols)
        v8f S0 = {bval,bval,bval,bval,bval,bval,bval,bval};
        {
            const float* Aly = sY + lh * C_DIM + 2*half;                 // A from LDS
            const float* B0  = Wy + (size_t)(cblk + wave*16 + lh) * C_DIM + 2*half;
            #pragma unroll 8
            for (int k0 = 0; k0 < C_DIM; k0 += 4) {
                v2f a  = *(const v2f*)(Aly + k0);
                v2f b0 = *(const v2f*)(B0 + k0);
                S0 = __builtin_amdgcn_wmma_f32_16x16x4_f32(false, a, false, b0, (short)0, S0, false, false);
            }
        }
        #pragma unroll
        for (int j = 0; j < 8; ++j)
            sS[(i * TM + j + 8*half) * CT + wave*16 + lh] = S0[j];
    }
    __syncthreads();

    // ---------------- Step 3: softmax over i (per (m,c)), write alpha ------------
    for (int p = tid; p < TM * CT; p += 256) {
        const int c = p & (CT - 1);
        const int m = p >> 7;
        float s[N_DIM];
        float mx = -3.4e38f;
        #pragma unroll
        for (int i = 0; i < N_DIM; ++i) {
            s[i] = sS[(i * TM + m) * CT + c];
            mx = fmaxf(mx, s[i]);
        }
        float sum = 0.f;
        #pragma unroll
        for (int i = 0; i < N_DIM; ++i) { s[i] = __expf(s[i] - mx); sum += s[i]; }
        const float inv = 1.f / sum;
        #pragma unroll
        for (int i = 0; i < N_DIM; ++i) {
            const float a = s[i] * inv;
            sS[(i * TM + m) * CT + c] = a;   // keep alpha in LDS for step 4
            alpha_out[((size_t)i * M_DIM + m0 + m) * C_DIM + cblk + c] = a;
        }
    }
    __syncthreads();

    // ---------------- Step 4: z[n,m,c] = sum_i alpha[i,m,c] * g[n,i,c] -----------
    for (int n = 0; n < N_DIM; ++n) {
        // stage g[n, 0:16, cblk:cblk+CT] into sY (8 KB, L2-hot slab)
        for (int q = tid; q < N_DIM * CT; q += 256) {
            const int gi = q >> 7;
            const int gc = q & (CT - 1);
            sY[gi * CT + gc] = g[((size_t)n * M_DIM + gi) * C_DIM + cblk + gc];
        }
        __syncthreads();
        for (int p = tid; p < TM * CT; p += 256) {
            const int c = p & (CT - 1);
            const int m = p >> 7;
            float acc = 0.f;
            #pragma unroll
            for (int i = 0; i < N_DIM; ++i)
                acc += sS[(i * TM + m) * CT + c] * sY[i * CT + c];
            z_out[((size_t)n * M_DIM + m0 + m) * C_DIM + cblk + c] = acc;
        }
        __syncthreads();
    }
}

extern "C" void kernel_launch(void* const* d_in, const int* in_sizes, int n_in,
                              void* d_out, int out_size, void* d_ws, size_t ws_size,
                              hipStream_t stream) {
    (void)in_sizes; (void)n_in; (void)d_ws; (void)ws_size; (void)out_size;
    const float* g  = (const float*)d_in[0];
    const float* h  = (const float*)d_in[1];
    const float* Wg = (const float*)d_in[2];
    const float* Wh = (const float*)d_in[3];
    const float* Wy = (const float*)d_in[4];
    const float* by = (const float*)d_in[5];
    float* alpha = (float*)d_out;
    float* z     = alpha + (size_t)N_DIM * M_DIM * C_DIM;
    dim3 grid(M_DIM / TM, C_DIM / CT);
    soft_attention_fused<<<grid, dim3(256), 0, stream>>>(g, h, Wg, Wh, Wy, by, alpha, z);
}